// GCN_78357383349033
// MI455X (gfx1250) — compile-verified
//
#include <hip/hip_runtime.h>

#define N_NODES 50000
#define N_EDGES 800000
#define NHID    128
#define NCLASS  40
#define RT      5      // row tiles (16 rows each) per block: 80 rows; 50000 = 625*80

typedef __attribute__((ext_vector_type(2))) float v2f;
typedef __attribute__((ext_vector_type(8))) float v8f;

__device__ __forceinline__ unsigned wang_hash(unsigned s) {
  s = (s ^ 61u) ^ (s >> 16);
  s *= 9u;
  s ^= s >> 4;
  s *= 0x27d4eb2du;
  s ^= s >> 15;
  return s;
}

__global__ __launch_bounds__(256) void zero_f32(float* __restrict__ p, int n) {
  int i = blockIdx.x * 256 + threadIdx.x;
  if (i < n) p[i] = 0.0f;
}

__global__ __launch_bounds__(256) void degree_kernel(const int* __restrict__ ed,
                                                     float* __restrict__ deg, int E) {
  int e = blockIdx.x * 256 + threadIdx.x;
  if (e < E) atomicAdd(&deg[ed[e]], 1.0f);
}

__global__ __launch_bounds__(256) void invdeg_kernel(float* __restrict__ deg, int n) {
  int i = blockIdx.x * 256 + threadIdx.x;
  if (i < n) deg[i] = 1.0f / fmaxf(deg[i], 1.0f);
}

// One wave32 per edge; each lane moves 4 contiguous floats (float4 load + 4 f32 atomics).
__global__ __launch_bounds__(256) void scatter_kernel(const float* __restrict__ feat,
                                                      const int* __restrict__ es,
                                                      const int* __restrict__ ed,
                                                      float* __restrict__ agg, int E) {
  int wid  = (blockIdx.x * 256 + threadIdx.x) >> 5;
  int lane = threadIdx.x & 31;
  if (wid >= E) return;
  int s = es[wid];
  int d = ed[wid];
  const float4 v = ((const float4*)(feat + (size_t)s * NHID))[lane];
  float* dst = agg + (size_t)d * NHID + lane * 4;
  atomicAdd(dst + 0, v.x);
  atomicAdd(dst + 1, v.y);
  atomicAdd(dst + 2, v.z);
  atomicAdd(dst + 3, v.w);
}

// out[N,128] = concat(h, agg*invdeg)[N,256] @ W[256,128] + b  (+ optional relu+dropout)
// Wave = one 16-col tile x RT row tiles (5 independent accumulators reuse each B frag).
// FP32 WMMA (V_WMMA_F32_16X16X4_F32) keeps full fp32 numerics.
__global__ __launch_bounds__(256) void sage_gemm_wmma(
    const float* __restrict__ h, const float* __restrict__ agg,
    const float* __restrict__ invdeg, const float* __restrict__ W,
    const float* __restrict__ bias, float* __restrict__ out,
    int relu_dropout)
{
  const int lane    = threadIdx.x & 31;
  const int wave    = threadIdx.x >> 5;       // 0..7 -> col tile
  const int colBase = wave * 16;
  const int half    = lane >> 4;              // K sub-block selector for A/B frags
  const int l15     = lane & 15;
  const int rowBase = blockIdx.x * (16 * RT);

  const float* hrow[RT];
  const float* grow[RT];
  float idg[RT];
  #pragma unroll
  for (int t = 0; t < RT; ++t) {
    int ar   = rowBase + t * 16 + l15;
    hrow[t]  = h   + (size_t)ar * NHID;
    grow[t]  = agg + (size_t)ar * NHID;
    idg[t]   = invdeg[ar];
  }

  v8f c[RT];
  #pragma unroll
  for (int t = 0; t < RT; ++t) c[t] = (v8f){};

  // Self-feature half of concat: K = 0..127
  for (int k0 = 0; k0 < NHID; k0 += 4) {
    int ka = k0 + half * 2;
    v2f b;
    b.x = W[(size_t)(ka + 0) * NHID + colBase + l15];
    b.y = W[(size_t)(ka + 1) * NHID + colBase + l15];
    #pragma unroll
    for (int t = 0; t < RT; ++t) {
      v2f a; a.x = hrow[t][ka]; a.y = hrow[t][ka + 1];
      c[t] = __builtin_amdgcn_wmma_f32_16x16x4_f32(false, a, false, b, (short)0, c[t], false, false);
    }
  }
  // Neighbor-mean half of concat: K = 128..255 (mean fused via invdeg scale)
  for (int k0 = 0; k0 < NHID; k0 += 4) {
    int ka = k0 + half * 2;
    v2f b;
    b.x = W[(size_t)(NHID + ka + 0) * NHID + colBase + l15];
    b.y = W[(size_t)(NHID + ka + 1) * NHID + colBase + l15];
    #pragma unroll
    for (int t = 0; t < RT; ++t) {
      v2f a; a.x = grow[t][ka] * idg[t]; a.y = grow[t][ka + 1] * idg[t];
      c[t] = __builtin_amdgcn_wmma_f32_16x16x4_f32(false, a, false, b, (short)0, c[t], false, false);
    }
  }

  const int ocol   = colBase + l15;
  const float bcol = bias[ocol];
  #pragma unroll
  for (int t = 0; t < RT; ++t) {
    #pragma unroll
    for (int r = 0; r < 8; ++r) {
      int orow = rowBase + t * 16 + r + half * 8; // C/D layout: lanes 16-31 hold M = r+8
      float v = c[t][r] + bcol;
      if (relu_dropout) {
        v = v > 0.0f ? v : 0.0f;
        unsigned hsh = wang_hash((unsigned)(orow * NHID + ocol) ^ 0x2545F491u);
        v = (hsh & 1u) ? v * 2.0f : 0.0f;       // keep=0.5 -> scale by 2
      }
      out[(size_t)orow * NHID + ocol] = v;
    }
  }
}

// logits[N,40] = h2[N,128] @ Wf[128,40] + bf.  3 waves cover 48 padded cols;
// B loads clamped+zeroed for col>=40 so EXEC stays uniform through WMMA.
__global__ __launch_bounds__(96) void final_gemm_wmma(
    const float* __restrict__ h, const float* __restrict__ W,
    const float* __restrict__ bias, float* __restrict__ out)
{
  const int lane    = threadIdx.x & 31;
  const int wave    = threadIdx.x >> 5;       // 0..2
  const int colBase = wave * 16;
  const int half    = lane >> 4;
  const int l15     = lane & 15;
  const int rowBase = blockIdx.x * (16 * RT);
  const int ocol    = colBase + l15;
  const int  cc     = ocol < NCLASS ? ocol : 0;
  const float cmask = ocol < NCLASS ? 1.0f : 0.0f;

  const float* hrow[RT];
  #pragma unroll
  for (int t = 0; t < RT; ++t)
    hrow[t] = h + (size_t)(rowBase + t * 16 + l15) * NHID;

  v8f c[RT];
  #pragma unroll
  for (int t = 0; t < RT; ++t) c[t] = (v8f){};

  for (int k0 = 0; k0 < NHID; k0 += 4) {
    int ka = k0 + half * 2;
    v2f b;
    b.x = W[(size_t)(ka + 0) * NCLASS + cc] * cmask;
    b.y = W[(size_t)(ka + 1) * NCLASS + cc] * cmask;
    #pragma unroll
    for (int t = 0; t < RT; ++t) {
      v2f a; a.x = hrow[t][ka]; a.y = hrow[t][ka + 1];
      c[t] = __builtin_amdgcn_wmma_f32_16x16x4_f32(false, a, false, b, (short)0, c[t], false, false);
    }
  }

  if (ocol >= NCLASS) return;
  const float bcol = bias[ocol];
  #pragma unroll
  for (int t = 0; t < RT; ++t) {
    #pragma unroll
    for (int r = 0; r < 8; ++r) {
      int orow = rowBase + t * 16 + r + half * 8;
      out[(size_t)orow * NCLASS + ocol] = c[t][r] + bcol;
    }
  }
}

__global__ __launch_bounds__(256) void log_softmax_kernel(float* __restrict__ out, int n) {
  int row = blockIdx.x * 256 + threadIdx.x;
  if (row >= n) return;
  float* p = out + (size_t)row * NCLASS;
  float m = -3.402823466e38f;
  for (int i = 0; i < NCLASS; ++i) m = fmaxf(m, p[i]);
  float s = 0.0f;
  for (int i = 0; i < NCLASS; ++i) s += __expf(p[i] - m);
  float lse = m + __logf(s);
  for (int i = 0; i < NCLASS; ++i) p[i] -= lse;
}

extern "C" void kernel_launch(void* const* d_in, const int* in_sizes, int n_in,
                              void* d_out, int out_size, void* d_ws, size_t ws_size,
                              hipStream_t stream) {
  const float* x  = (const float*)d_in[0];
  const int*   es = (const int*)d_in[1];
  const int*   ed = (const int*)d_in[2];
  const float* W1 = (const float*)d_in[3];
  const float* b1 = (const float*)d_in[4];
  const float* W2 = (const float*)d_in[5];
  const float* b2 = (const float*)d_in[6];
  const float* Wf = (const float*)d_in[7];
  const float* bf = (const float*)d_in[8];
  float* out = (float*)d_out;
  (void)in_sizes; (void)n_in; (void)out_size; (void)ws_size;

  // Workspace layout (floats): deg[N] | agg[N*128] | h1[N*128] | h2[N*128]  (~77 MB)
  float* ws  = (float*)d_ws;
  float* deg = ws;
  float* agg = deg + N_NODES;
  float* h1  = agg + (size_t)N_NODES * NHID;
  float* h2  = h1  + (size_t)N_NODES * NHID;

  const int nAgg      = N_NODES * NHID;
  const int gemmGrid  = N_NODES / (16 * RT);   // 625
  dim3 b256(256);

  // ---- Layer 1 ----
  zero_f32<<<(N_NODES + 255) / 256, b256, 0, stream>>>(deg, N_NODES);
  zero_f32<<<(nAgg + 255) / 256, b256, 0, stream>>>(agg, nAgg);
  degree_kernel<<<(N_EDGES + 255) / 256, b256, 0, stream>>>(ed, deg, N_EDGES);
  scatter_kernel<<<N_EDGES / 8, b256, 0, stream>>>(x, es, ed, agg, N_EDGES);
  invdeg_kernel<<<(N_NODES + 255) / 256, b256, 0, stream>>>(deg, N_NODES);
  sage_gemm_wmma<<<gemmGrid, b256, 0, stream>>>(x, agg, deg, W1, b1, h1, 1);

  // ---- Layer 2 ----
  zero_f32<<<(nAgg + 255) / 256, b256, 0, stream>>>(agg, nAgg);
  scatter_kernel<<<N_EDGES / 8, b256, 0, stream>>>(h1, es, ed, agg, N_EDGES);
  sage_gemm_wmma<<<gemmGrid, b256, 0, stream>>>(h1, agg, deg, W2, b2, h2, 0);

  // ---- Classifier + log-softmax ----
  final_gemm_wmma<<<gemmGrid, dim3(96), 0, stream>>>(h2, Wf, bf, out);
  log_softmax_kernel<<<(N_NODES + 255) / 256, b256, 0, stream>>>(out, N_NODES);
}